// Network_24962349925018
// MI455X (gfx1250) — compile-verified
//
#include <hip/hip_runtime.h>
#include <hip/hip_bf16.h>

typedef __attribute__((ext_vector_type(2))) float v2f;
typedef __attribute__((ext_vector_type(8))) float v8f;

#define FEAT_DIM 360
#define TW 32
#define TH 32

__device__ __forceinline__ float lrelu(float v) {
    return v >= 0.f ? v : 0.2f * v;
}

// ---------------------------------------------------------------------------
// Kernel 1: per-(sample,channel) mean of thumb_x. 24 blocks x 256 threads.
// thumb: (8,3,64,64) contiguous -> channel bc occupies 4096 contiguous floats.
// ---------------------------------------------------------------------------
__global__ __launch_bounds__(256) void mean_kernel(const float* __restrict__ thumb,
                                                   float* __restrict__ means) {
    __shared__ float red[256];
    const int bc = blockIdx.x;                 // 0..23 = b*3+c
    const float* p = thumb + (size_t)bc * 4096;
    float s = 0.f;
    for (int i = threadIdx.x; i < 4096; i += 256) s += p[i];
    red[threadIdx.x] = s;
    __syncthreads();
    for (int off = 128; off > 0; off >>= 1) {
        if (threadIdx.x < off) red[threadIdx.x] += red[threadIdx.x + off];
        __syncthreads();
    }
    if (threadIdx.x == 0) means[bc] = red[0] * (1.f / 4096.f);
}

// ---------------------------------------------------------------------------
// Kernel 2: feature = means(8x3) @ Wm(3x360) + bm  via V_WMMA_F32_16X16X4_F32.
// One wave32. A = means padded to 16x4 (M=batch,K=channel), B = Wm tile 4x16,
// C = bias broadcast per column. 23 N-tiles cover 360 columns.
//
// A layout (16x4 f32): lane L: m=L&15, VGPR0 = A[m][2*(L>>4)], VGPR1 = A[m][2*(L>>4)+1]
// B layout (4x16 f32): lane L: n=L&15, VGPR0 = B[2*(L>>4)][n], VGPR1 = B[2*(L>>4)+1][n]
// C/D layout (16x16): VGPR r: lanes0-15 -> (M=r, N=lane); lanes16-31 -> (M=r+8, N=lane-16)
// ---------------------------------------------------------------------------
__global__ __launch_bounds__(32) void feature_wmma_kernel(const float* __restrict__ means,
                                                          const float* __restrict__ Wm,
                                                          const float* __restrict__ bm,
                                                          float* __restrict__ featOut) {
    const int lane  = threadIdx.x;
    const int m     = lane & 15;
    const int khalf = lane >> 4;
    const int k0    = 2 * khalf;

    v2f a;
    a.x = (m < 8 && (k0 + 0) < 3) ? means[m * 3 + k0 + 0] : 0.f;
    a.y = (m < 8 && (k0 + 1) < 3) ? means[m * 3 + k0 + 1] : 0.f;

    for (int t = 0; t < 23; ++t) {
        const int fcol = t * 16 + m;          // column index (N) for this lane
        const bool fok = fcol < FEAT_DIM;

        v2f b;
        b.x = (fok && (k0 + 0) < 3) ? Wm[(k0 + 0) * FEAT_DIM + fcol] : 0.f;
        b.y = (fok && (k0 + 1) < 3) ? Wm[(k0 + 1) * FEAT_DIM + fcol] : 0.f;

        const float bias = fok ? bm[fcol] : 0.f;
        v8f c;
#pragma unroll
        for (int r = 0; r < 8; ++r) c[r] = bias;

        v8f d = __builtin_amdgcn_wmma_f32_16x16x4_f32(
            /*neg_a=*/false, a, /*neg_b=*/false, b,
            /*c_mod=*/(short)0, c, /*reuse_a=*/false, /*reuse_b=*/false);

        if (lane < 16 && fok) {
#pragma unroll
            for (int r = 0; r < 8; ++r)       // rows M=0..7 live in lanes 0..15
                featOut[r * FEAT_DIM + fcol] = d[r];
        }
    }
}

// ---------------------------------------------------------------------------
// Kernel 3: fused conv_stack + attention on one 32x32 output tile per block.
// All three 3x3 layers chained through LDS with proper SAME-padding semantics
// (outputs outside the image are forced to 0 before feeding the next layer),
// then 5x conv1x1 + residual + polynomial attention in registers.
// ---------------------------------------------------------------------------
__global__ __launch_bounds__(256) void conv_att_kernel(const float* __restrict__ xin,
                                                       const float* __restrict__ feat,
                                                       float* __restrict__ out,
                                                       int H, int W) {
    const int ox  = blockIdx.x * TW;
    const int oy  = blockIdx.y * TH;
    const int bidx = blockIdx.z;
    const int tid = threadIdx.x;

    __shared__ float sF[FEAT_DIM];
    __shared__ float bufIn[3][TH + 6][TW + 6];   // input, halo 3
    __shared__ float buf1[3][TH + 4][TW + 4];    // after layer 1, halo 2
    __shared__ float buf2[3][TH + 2][TW + 2];    // after layer 2, halo 1

    const float* fsrc = feat + bidx * FEAT_DIM;
    for (int i = tid; i < FEAT_DIM; i += 256) sF[i] = fsrc[i];

    // ---- stage input tile with halo 3 (zero-padded at image border) ----
    const float* src = xin + (size_t)bidx * 3 * H * W;
    const int NIN = 3 * (TH + 6) * (TW + 6);
    for (int i = tid; i < NIN; i += 256) {
        const int c   = i / ((TH + 6) * (TW + 6));
        const int rem = i % ((TH + 6) * (TW + 6));
        const int r   = rem / (TW + 6);
        const int q   = rem % (TW + 6);
        const int y = oy - 3 + r, x = ox - 3 + q;
        float v = 0.f;
        if (y >= 0 && y < H && x >= 0 && x < W)
            v = src[(size_t)c * H * W + (size_t)y * W + x];
        bufIn[c][r][q] = v;
    }
    __syncthreads();

    // ---- conv3x3 layer 1: (TH+6)x(TW+6) -> (TH+4)x(TW+4) ----
    {
        const float* wf = sF;                 // 81 weights + 3 bias
        const int NP = (TH + 4) * (TW + 4);
        for (int i = tid; i < NP; i += 256) {
            const int r = i / (TW + 4), q = i % (TW + 4);
            const int y = oy - 2 + r, x = ox - 2 + q;
            const bool inside = (y >= 0 && y < H && x >= 0 && x < W);
            float a0 = wf[81], a1 = wf[82], a2 = wf[83];
#pragma unroll
            for (int ci = 0; ci < 3; ++ci)
#pragma unroll
                for (int kh = 0; kh < 3; ++kh)
#pragma unroll
                    for (int kw = 0; kw < 3; ++kw) {
                        const float v = bufIn[ci][r + kh][q + kw];
                        a0 = fmaf(wf[0 * 27 + ci * 9 + kh * 3 + kw], v, a0);
                        a1 = fmaf(wf[1 * 27 + ci * 9 + kh * 3 + kw], v, a1);
                        a2 = fmaf(wf[2 * 27 + ci * 9 + kh * 3 + kw], v, a2);
                    }
            buf1[0][r][q] = inside ? lrelu(a0) : 0.f;
            buf1[1][r][q] = inside ? lrelu(a1) : 0.f;
            buf1[2][r][q] = inside ? lrelu(a2) : 0.f;
        }
    }
    __syncthreads();

    // ---- conv3x3 layer 2: (TH+4)x(TW+4) -> (TH+2)x(TW+2) ----
    {
        const float* wf = sF + 84;
        const int NP = (TH + 2) * (TW + 2);
        for (int i = tid; i < NP; i += 256) {
            const int r = i / (TW + 2), q = i % (TW + 2);
            const int y = oy - 1 + r, x = ox - 1 + q;
            const bool inside = (y >= 0 && y < H && x >= 0 && x < W);
            float a0 = wf[81], a1 = wf[82], a2 = wf[83];
#pragma unroll
            for (int ci = 0; ci < 3; ++ci)
#pragma unroll
                for (int kh = 0; kh < 3; ++kh)
#pragma unroll
                    for (int kw = 0; kw < 3; ++kw) {
                        const float v = buf1[ci][r + kh][q + kw];
                        a0 = fmaf(wf[0 * 27 + ci * 9 + kh * 3 + kw], v, a0);
                        a1 = fmaf(wf[1 * 27 + ci * 9 + kh * 3 + kw], v, a1);
                        a2 = fmaf(wf[2 * 27 + ci * 9 + kh * 3 + kw], v, a2);
                    }
            buf2[0][r][q] = inside ? lrelu(a0) : 0.f;
            buf2[1][r][q] = inside ? lrelu(a1) : 0.f;
            buf2[2][r][q] = inside ? lrelu(a2) : 0.f;
        }
    }
    __syncthreads();

    // ---- conv3x3 layer 3 + 5x conv1x1 + residual + attention, in registers ----
    const int NPIX = TH * TW;
    for (int i = tid; i < NPIX; i += 256) {
        const int r = i / TW, q = i % TW;
        const int y = oy + r, x = ox + q;
        const float* wf = sF + 168;
        float v0 = wf[81], v1 = wf[82], v2 = wf[83];
#pragma unroll
        for (int ci = 0; ci < 3; ++ci)
#pragma unroll
            for (int kh = 0; kh < 3; ++kh)
#pragma unroll
                for (int kw = 0; kw < 3; ++kw) {
                    const float v = buf2[ci][r + kh][q + kw];
                    v0 = fmaf(wf[0 * 27 + ci * 9 + kh * 3 + kw], v, v0);
                    v1 = fmaf(wf[1 * 27 + ci * 9 + kh * 3 + kw], v, v1);
                    v2 = fmaf(wf[2 * 27 + ci * 9 + kh * 3 + kw], v, v2);
                }
        v0 = lrelu(v0); v1 = lrelu(v1); v2 = lrelu(v2);

        // 5x conv1x1: k[o][i] = wc[o*3+i], bias wc[9+o]
#pragma unroll
        for (int l = 0; l < 5; ++l) {
            const float* wc = sF + 252 + l * 12;
            const float u0 = fmaf(wc[0], v0, fmaf(wc[1], v1, fmaf(wc[2], v2, wc[9])));
            const float u1 = fmaf(wc[3], v0, fmaf(wc[4], v1, fmaf(wc[5], v2, wc[10])));
            const float u2 = fmaf(wc[6], v0, fmaf(wc[7], v1, fmaf(wc[8], v2, wc[11])));
            v0 = lrelu(u0); v1 = lrelu(u1); v2 = lrelu(u2);
        }

        // residual (original input at center of halo-3 buffer)
        float vv0 = v0 + bufIn[0][r + 3][q + 3];
        float vv1 = v1 + bufIn[1][r + 3][q + 3];
        float vv2 = v2 + bufIn[2][r + 3][q + 3];

        // polynomial attention on the post-conv values
        const float hm = (float)y / (float)H;
        const float wm = (float)x / (float)W;
        float vv[3] = {vv0, vv1, vv2};
#pragma unroll
        for (int c = 0; c < 3; ++c) {
            const float* pa = sF + 312 + c * 16;
            float att = 1.f;
#pragma unroll
            for (int p = 0; p < 4; ++p)
                att *= fmaf(pa[p * 4 + 0], hm,
                       fmaf(pa[p * 4 + 1], wm,
                       fmaf(pa[p * 4 + 2], vv[c], pa[p * 4 + 3])));
            vv[c] = vv[c] * (1.f + att);
        }

        const size_t base = (size_t)bidx * 3 * H * W + (size_t)y * W + x;
        out[base]                     = vv[0];
        out[base + (size_t)H * W]     = vv[1];
        out[base + 2 * (size_t)H * W] = vv[2];
    }
}

// ---------------------------------------------------------------------------
extern "C" void kernel_launch(void* const* d_in, const int* in_sizes, int n_in,
                              void* d_out, int out_size, void* d_ws, size_t ws_size,
                              hipStream_t stream) {
    (void)in_sizes; (void)n_in; (void)out_size; (void)ws_size;

    const float* x     = (const float*)d_in[0];   // (8,3,512,512)
    const float* thumb = (const float*)d_in[1];   // (8,3,64,64)
    const float* Wm    = (const float*)d_in[2];   // (3,360)
    const float* bm    = (const float*)d_in[3];   // (360,)
    float* out = (float*)d_out;                   // x_out then thumb_out, flat
    float* ws  = (float*)d_ws;

    float* means = ws;        // 24 floats
    float* feat  = ws + 32;   // 8*360 floats

    mean_kernel<<<24, 256, 0, stream>>>(thumb, means);
    feature_wmma_kernel<<<1, 32, 0, stream>>>(means, Wm, bm, feat);

    dim3 gx(512 / TW, 512 / TH, 8);
    conv_att_kernel<<<gx, 256, 0, stream>>>(x, feat, out, 512, 512);

    dim3 gt(64 / TW, 64 / TH, 8);
    conv_att_kernel<<<gt, 256, 0, stream>>>(thumb, feat,
                                            out + (size_t)8 * 3 * 512 * 512, 64, 64);
}